// Aggregator_87935160418875
// MI455X (gfx1250) — compile-verified
//
#include <hip/hip_runtime.h>

#define N 1024
#define H 256

typedef __attribute__((ext_vector_type(2))) float v2f;
typedef __attribute__((ext_vector_type(8))) float v8f;

// ---------------------------------------------------------------- small prep
__global__ void prep_pb(const float* __restrict__ ip, const float* __restrict__ skip,
                        const float* __restrict__ bd,
                        float* __restrict__ pbt, float* __restrict__ pbf) {
  int i = blockIdx.x * blockDim.x + threadIdx.x;
  if (i < N) {
    float w = ip[i] * skip[(size_t)i * N + i];   // ip * no_skip
    pbt[i] = w * bd[2 * i];
    pbf[i] = w * bd[2 * i + 1];
  }
}

__global__ void zero_f32(float* __restrict__ p, int n) {
  int i = blockIdx.x * blockDim.x + threadIdx.x;
  if (i < n) p[i] = 0.0f;
}

// M[i, ti[i]] += pbt[i]; M[i, fi[i]] += pbf[i]  (row i owned by thread i -> no atomics)
__global__ void scatter_M(const int* __restrict__ ti, const int* __restrict__ fi,
                          const float* __restrict__ pbt, const float* __restrict__ pbf,
                          float* __restrict__ M) {
  int i = blockIdx.x * blockDim.x + threadIdx.x;
  if (i < N) {
    int t = ti[i], f = fi[i];
    if (t == f) {
      M[(size_t)i * N + t] = pbt[i] + pbf[i];
    } else {
      M[(size_t)i * N + t] = pbt[i];
      M[(size_t)i * N + f] = pbf[i];
    }
  }
}

// wT[j][i] = ip[i] * skip[i][j], diag zeroed. LDS-tiled transpose, both sides coalesced.
__global__ void build_wT(const float* __restrict__ ip, const float* __restrict__ skip,
                         float* __restrict__ wT) {
  __shared__ float tile[32][33];
  int j0 = blockIdx.x * 32;
  int i0 = blockIdx.y * 32;
  int tx = threadIdx.x, ty = threadIdx.y;
  #pragma unroll
  for (int r = 0; r < 32; r += 8) {
    int i = i0 + ty + r;
    int j = j0 + tx;
    float v = ip[i] * skip[(size_t)i * N + j];
    if (i == j) v = 0.0f;
    tile[ty + r][tx] = v;
  }
  __syncthreads();
  #pragma unroll
  for (int r = 0; r < 32; r += 8) {
    int j = j0 + ty + r;
    int i = i0 + tx;
    wT[(size_t)j * N + i] = tile[tx][ty + r];
  }
}

// new_ip[j] = sum_i wT[j][i] + sum_{ti[i]==j} pbt[i] + sum_{fi[i]==j} pbf[i]
__global__ void new_ip_kernel(const float* __restrict__ wT,
                              const int* __restrict__ ti, const int* __restrict__ fi,
                              const float* __restrict__ pbt, const float* __restrict__ pbf,
                              float* __restrict__ out_ip) {
  int j = blockIdx.x;
  int tid = threadIdx.x;
  float s = 0.0f;
  for (int i = tid; i < N; i += 256) {
    s += wT[(size_t)j * N + i];
    if (ti[i] == j) s += pbt[i];
    if (fi[i] == j) s += pbf[i];
  }
  __shared__ float red[256];
  red[tid] = s;
  __syncthreads();
  #pragma unroll
  for (int off = 128; off > 0; off >>= 1) {
    if (tid < off) red[tid] += red[tid + off];
    __syncthreads();
  }
  if (tid == 0) out_ip[j] = red[0];
}

// TF[j,h] = sum_i M[i,j] * hp[i,h]  — f32 WMMA 16x16x4, one wave per 16x16 tile.
// A-matrix (16x4): lanes 0-15 hold K=0,1 in v[0],v[1]; lanes 16-31 hold K=2,3.
// B-matrix (4x16): v[0]: lanes0-15 row K=0, lanes16-31 row K=1; v[1]: rows K=2,3.
// C/D (16x16): lane(0-15) n=lane, VGPR r -> M=r; lanes 16-31 -> M=8+r.
__global__ void __launch_bounds__(128) wmma_tf(const float* __restrict__ M,
                                               const float* __restrict__ hp,
                                               float* __restrict__ TF) {
  int wid  = blockIdx.x * 4 + (threadIdx.x >> 5);
  int lane = threadIdx.x & 31;
  int jb = (wid >> 4) << 4;   // 64 j-tiles
  int hb = (wid & 15) << 4;   // 16 h-tiles
  int half = lane >> 4;       // 0 or 1
  int l = lane & 15;
  v8f c = {0.f, 0.f, 0.f, 0.f, 0.f, 0.f, 0.f, 0.f};
  for (int k = 0; k < N; k += 4) {
    v2f a, b;
    int kk = half * 2;
    a.x = M[(size_t)(k + kk) * N + jb + l];        // A[m=l, kk]   = M[i=k+kk, j=jb+l]
    a.y = M[(size_t)(k + kk + 1) * N + jb + l];
    b.x = hp[(size_t)(k + half) * H + hb + l];     // B[kk=half,   n=l]
    b.y = hp[(size_t)(k + 2 + half) * H + hb + l]; // B[kk=2+half, n=l]
    c = __builtin_amdgcn_wmma_f32_16x16x4_f32(false, a, false, b, (short)0, c,
                                              false, false);
  }
  #pragma unroll
  for (int r = 0; r < 8; ++r) {
    int m = r + half * 8;
    TF[(size_t)(jb + m) * H + hb + l] = c[r];
  }
}

// The bandwidth kernel: out_hidden[j,h] = (TF[j,h] + sum_i wrow[i]*hsp[i,j,h]) / (ip[j]+1e-7)
// Block per j. 256 threads = 4 i-partitions x 64 float4-lanes over H=256.
__global__ void __launch_bounds__(256) hidden_kernel(const float* __restrict__ hsp,
                                                     const float* __restrict__ wT,
                                                     const float* __restrict__ TF,
                                                     const float* __restrict__ out_ip,
                                                     float* __restrict__ out_hidden) {
  int j = blockIdx.x;
  int tid = threadIdx.x;
  __shared__ float wrow[N];
  __shared__ float4 accbuf[256];

  for (int i = tid; i < N; i += 256) wrow[i] = wT[(size_t)j * N + i];
  __syncthreads();

  const int q = tid >> 6;        // i-partition 0..3
  const int t = tid & 63;        // float4 lane over H
  const int I0 = q * 256;
  const float* base = hsp + (size_t)j * H + (size_t)t * 4;

  float ax = 0.f, ay = 0.f, az = 0.f, aw = 0.f;
  #pragma unroll 8
  for (int ii = 0; ii < 256; ++ii) {
    int i = I0 + ii;
    const float4 v = *reinterpret_cast<const float4*>(base + (size_t)i * ((size_t)N * H));
    if (ii < 248)
      __builtin_prefetch(base + (size_t)(i + 8) * ((size_t)N * H), 0, 1);
    float w = wrow[i];
    ax = fmaf(w, v.x, ax);
    ay = fmaf(w, v.y, ay);
    az = fmaf(w, v.z, az);
    aw = fmaf(w, v.w, aw);
  }
  accbuf[tid] = make_float4(ax, ay, az, aw);
  __syncthreads();

  if (tid < 64) {
    float4 s0 = accbuf[tid], s1 = accbuf[tid + 64], s2 = accbuf[tid + 128], s3 = accbuf[tid + 192];
    float4 tf = *reinterpret_cast<const float4*>(TF + (size_t)j * H + (size_t)t * 4);
    float inv = 1.0f / (out_ip[j] + 1e-7f);
    float4 o;
    o.x = (s0.x + s1.x + s2.x + s3.x + tf.x) * inv;
    o.y = (s0.y + s1.y + s2.y + s3.y + tf.y) * inv;
    o.z = (s0.z + s1.z + s2.z + s3.z + tf.z) * inv;
    o.w = (s0.w + s1.w + s2.w + s3.w + tf.w) * inv;
    *reinterpret_cast<float4*>(out_hidden + (size_t)j * H + (size_t)t * 4) = o;
  }
}

// ---------------------------------------------------------------- launcher
extern "C" void kernel_launch(void* const* d_in, const int* in_sizes, int n_in,
                              void* d_out, int out_size, void* d_ws, size_t ws_size,
                              hipStream_t stream) {
  const float* ip   = (const float*)d_in[0];   // (N,)
  const float* hp   = (const float*)d_in[1];   // (N,H)
  const float* hsp  = (const float*)d_in[2];   // (N,N,H)
  const float* skip = (const float*)d_in[3];   // (N,N)
  const float* bd   = (const float*)d_in[4];   // (N,2)
  const int*   ti   = (const int*)d_in[5];     // (N,)
  const int*   fi   = (const int*)d_in[6];     // (N,)

  float* out    = (float*)d_out;               // [N] ip ++ [N*H] hidden
  float* ws     = (float*)d_ws;
  float* wT     = ws;                          // N*N
  float* Mmat   = ws + (size_t)N * N;          // N*N
  float* pbt    = ws + 2 * (size_t)N * N;      // N
  float* pbf    = pbt + N;                     // N
  float* TF     = pbf + N;                     // N*H

  prep_pb<<<(N + 255) / 256, 256, 0, stream>>>(ip, skip, bd, pbt, pbf);
  zero_f32<<<(N * N) / 256, 256, 0, stream>>>(Mmat, N * N);
  scatter_M<<<(N + 255) / 256, 256, 0, stream>>>(ti, fi, pbt, pbf, Mmat);
  build_wT<<<dim3(N / 32, N / 32), dim3(32, 8), 0, stream>>>(ip, skip, wT);
  new_ip_kernel<<<N, 256, 0, stream>>>(wT, ti, fi, pbt, pbf, out);
  wmma_tf<<<(N / 16) * (H / 16) / 4, 128, 0, stream>>>(Mmat, hp, TF);
  hidden_kernel<<<N, 256, 0, stream>>>(hsp, wT, TF, out, out + N);
}